// SpikeFP8Multiplier_76312978916090
// MI455X (gfx1250) — compile-verified
//
#include <hip/hip_runtime.h>

typedef unsigned int       u32;
typedef unsigned long long u64;
typedef u32 u32x4 __attribute__((ext_vector_type(4)));

#define ONE_F32 0x3F800000u
#define BIAS 7

// round-to-nearest-even right shift of non-negative M by sh (sh >= 0)
__device__ __forceinline__ int rne_rshift(int M, int sh) {
    int q    = M >> sh;
    int rem  = M - (q << sh);
    int half = (1 << sh) >> 1;                 // 0 when sh==0
    int up   = (int)(((rem > half) || ((rem == half) && (q & 1))) && (sh > 0));
    return q + up;
}

__global__ __launch_bounds__(256)
void fp8_bitmul_kernel(const u32* __restrict__ a, const u32* __restrict__ b,
                       u32* __restrict__ out, int n) {
    __shared__ u32x4 shA[512];   // 8 KB: two u32x4 per thread
    __shared__ u32x4 shB[512];   // 8 KB

    const int tid = threadIdx.x;
    const int idx = blockIdx.x * 256 + tid;
    if (idx >= n) return;

    // ---- stage both inputs through LDS with gfx1250 async loads ----
    u64 gA   = (u64)(const void*)(a + (size_t)idx * 8);
    u64 gB   = (u64)(const void*)(b + (size_t)idx * 8);
    u32 ldsA = (u32)(u64)(&shA[2 * tid]);   // flat shared ptr low 32 bits == LDS offset
    u32 ldsB = (u32)(u64)(&shB[2 * tid]);

    asm volatile(
        "global_load_async_to_lds_b128 %0, %2, off\n\t"
        "global_load_async_to_lds_b128 %0, %2, off offset:16\n\t"
        "global_load_async_to_lds_b128 %1, %3, off\n\t"
        "global_load_async_to_lds_b128 %1, %3, off offset:16\n\t"
        "s_wait_asynccnt 0x0"
        :
        : "v"(ldsA), "v"(ldsB), "v"(gA), "v"(gB)
        : "memory");

    u32x4 a0 = shA[2 * tid + 0];
    u32x4 a1 = shA[2 * tid + 1];
    u32x4 b0 = shB[2 * tid + 0];
    u32x4 b1 = shB[2 * tid + 1];

    // ---- decode: each word is 0x0 (0.0f) or 0x3F800000 (1.0f); bit = word >> 29 ----
    u32 sa = a0.x >> 29;
    int ea = (int)(((a0.y >> 29) << 3) | ((a0.z >> 29) << 2) | ((a0.w >> 29) << 1) | (a1.x >> 29));
    int ma = (int)(((a1.y >> 29) << 2) | ((a1.z >> 29) << 1) | (a1.w >> 29));

    u32 sb = b0.x >> 29;
    int eb = (int)(((b0.y >> 29) << 3) | ((b0.z >> 29) << 2) | ((b0.w >> 29) << 1) | (b1.x >> 29));
    int mb = (int)(((b1.y >> 29) << 2) | ((b1.z >> 29) << 1) | (b1.w >> 29));

    // ---- exact FP8 E4M3 multiply (mirrors reference) ----
    int sig_a = (ea == 0) ? ma : (ma + 8);
    int sig_b = (eb == 0) ? mb : (mb + 8);
    int exa   = ((ea == 0) ? 1 : ea) - BIAS;
    int exb   = ((eb == 0) ? 1 : eb) - BIAS;
    int M = sig_a * sig_b;                 // exact product, <= 225
    int E = exa + exb - 6;                 // value = M * 2^E
    int p = (M != 0) ? (31 - (int)__builtin_clz((u32)M)) : 0;  // MSB position
    int eb_out = E + p + BIAS;

    // normal path (eb_out >= 1): RNE to 3 mantissa bits
    int sh    = p - 3;
    int shp   = (sh > 0) ? sh : 0;
    int lsh   = (sh < 0) ? -sh : 0;
    int mant4 = rne_rshift(M, shp) << lsh;     // in [8,16] when M!=0
    int carry = mant4 >> 4;                    // rounding overflow -> exp += 1
    mant4 = (mant4 == 16) ? 8 : mant4;
    int e_n = eb_out + carry;
    int m_n = mant4 - 8;
    if (e_n >= 16) { e_n = 15; m_n = 7; }      // overflow -> NaN (e=1111, m=111)

    // subnormal path (eb_out <= 0): mant = RNE(M * 2^(E+9))
    int t   = E + 9;
    int nsh = -t; nsh = (nsh < 0) ? 0 : ((nsh > 30) ? 30 : nsh);
    int pls = (t > 0) ? t : 0;
    int mant_s = rne_rshift(M, nsh) << pls;
    int e_s = (mant_s >= 8) ? 1 : 0;           // promoted to smallest normal
    int m_s = (mant_s >= 8) ? 0 : mant_s;

    bool is_sub = (eb_out <= 0);
    int e_o = is_sub ? e_s : e_n;
    int m_o = is_sub ? m_s : m_n;
    if (M == 0) { e_o = 0; m_o = 0; }          // true zero
    u32 s_o = sa ^ sb;

    // ---- encode bits back to 0.0f / 1.0f and stream out (non-temporal) ----
    u32x4 o0, o1;
    o0.x = s_o        ? ONE_F32 : 0u;
    o0.y = (e_o & 8)  ? ONE_F32 : 0u;
    o0.z = (e_o & 4)  ? ONE_F32 : 0u;
    o0.w = (e_o & 2)  ? ONE_F32 : 0u;
    o1.x = (e_o & 1)  ? ONE_F32 : 0u;
    o1.y = (m_o & 4)  ? ONE_F32 : 0u;
    o1.z = (m_o & 2)  ? ONE_F32 : 0u;
    o1.w = (m_o & 1)  ? ONE_F32 : 0u;

    u32x4* op = (u32x4*)(out + (size_t)idx * 8);
    __builtin_nontemporal_store(o0, op);
    __builtin_nontemporal_store(o1, op + 1);
}

extern "C" void kernel_launch(void* const* d_in, const int* in_sizes, int n_in,
                              void* d_out, int out_size, void* d_ws, size_t ws_size,
                              hipStream_t stream) {
    const u32* a = (const u32*)d_in[0];
    const u32* b = (const u32*)d_in[1];
    u32* out = (u32*)d_out;

    int n = in_sizes[0] / 8;                 // number of FP8 elements (1024*4096)
    int blocks = (n + 255) / 256;
    fp8_bitmul_kernel<<<blocks, 256, 0, stream>>>(a, b, out, n);
}